// GNNModel_58394375357177
// MI455X (gfx1250) — compile-verified
//
#include <hip/hip_runtime.h>
#include <hip/hip_bf16.h>

// ---------------------------------------------------------------------------
// GAT (2 layers, single head) for MI455X / gfx1250.
//  - Dense projections x@W1 and h@W2 use V_WMMA_F32_16X16X4_F32 (f32 WMMA,
//    full f32 precision vs the f32 reference).
//  - Edge softmax-aggregation via two edge-parallel passes with L2 atomics
//    (atomicMax on monotonic-uint-mapped floats, atomicAdd f32).
// ---------------------------------------------------------------------------

typedef __attribute__((ext_vector_type(2))) float v2f;
typedef __attribute__((ext_vector_type(8))) float v8f;

#define NODES   100000
#define LRELU   0.2f

// ---------------------------------------------------------------------------
// GEMM1: h1[N,16] = x[N,128] @ W1[128,16]   (one wave per 16-row tile)
// A frag (16x4 f32): lane<16 -> M=lane, K={k,k+1}; lane>=16 -> M=lane-16, K={k+2,k+3}
// B frag (4x16 f32): lane<16 -> N=lane, K={k,k+1}; lane>=16 -> N=lane-16, K={k+2,k+3}
// C/D (16x16 f32): VGPR j: lanes0-15 M=j, lanes16-31 M=j+8; N=lane&15
// ---------------------------------------------------------------------------
__global__ __launch_bounds__(128) void gemm_x_w1(const float* __restrict__ x,
                                                 const float* __restrict__ W1,
                                                 float* __restrict__ h1,
                                                 int nTiles) {
    __shared__ __align__(16) float sB[128 * 16];
    __shared__ __align__(16) float sA[4][16 * 132];   // pitch 132: no bank conflicts

    const int lane = threadIdx.x & 31;
    const int wave = threadIdx.x >> 5;
    const int tile = blockIdx.x * 4 + wave;

    // Stage W1 (128x16) into LDS, coalesced across the whole block.
    for (int i = threadIdx.x; i < 128 * 16; i += 128) sB[i] = W1[i];

    if (tile < nTiles) {
        // Stage this wave's 16x128 A tile: one row (512B) per iteration, float4/lane.
        const float* Ax = x + (size_t)tile * 16 * 128;
#pragma unroll 4
        for (int r = 0; r < 16; ++r) {
            float4 v = ((const float4*)(Ax + r * 128))[lane];
            *(float4*)&sA[wave][r * 132 + lane * 4] = v;
        }
    }
    __syncthreads();

    if (tile < nTiles) {
        v8f c = {0.f, 0.f, 0.f, 0.f, 0.f, 0.f, 0.f, 0.f};
        const int mrow = lane & 15;
        const int koff = (lane >> 4) * 2;
        const int ncol = lane & 15;
        const float* aRow = &sA[wave][mrow * 132];
#pragma unroll
        for (int k = 0; k < 128; k += 4) {
            v2f a, b;
            a.x = aRow[k + koff];
            a.y = aRow[k + koff + 1];
            b.x = sB[(k + koff) * 16 + ncol];
            b.y = sB[(k + koff + 1) * 16 + ncol];
            c = __builtin_amdgcn_wmma_f32_16x16x4_f32(false, a, false, b,
                                                      (short)0, c, false, false);
        }
        const int row0 = tile * 16 + (lane >> 4) * 8;
#pragma unroll
        for (int j = 0; j < 8; ++j)
            h1[(size_t)(row0 + j) * 16 + ncol] = c[j];
    }
}

// ---------------------------------------------------------------------------
// GEMM2: h2[N,8] = A[N,16] @ W2[16,8]  (W2 zero-padded to 16x16 in LDS)
// ---------------------------------------------------------------------------
__global__ __launch_bounds__(128) void gemm_h_w2(const float* __restrict__ A,
                                                 const float* __restrict__ W2,
                                                 float* __restrict__ h2,
                                                 int nTiles) {
    __shared__ __align__(16) float sB[16 * 16];
    __shared__ __align__(16) float sA[4][16 * 20];    // pitch 20: conflict-free

    const int lane = threadIdx.x & 31;
    const int wave = threadIdx.x >> 5;
    const int tile = blockIdx.x * 4 + wave;

    if (threadIdx.x < 256) {
        int k = threadIdx.x >> 4, n = threadIdx.x & 15;
        sB[threadIdx.x] = (n < 8) ? W2[k * 8 + n] : 0.f;
    }
    if (tile < nTiles) {
        const float* Ab = A + (size_t)tile * 16 * 16;   // 16 rows x 16 = contiguous
        for (int idx = lane; idx < 64; idx += 32) {
            float4 v = ((const float4*)Ab)[idx];
            int r = idx >> 2, cc = (idx & 3) * 4;
            *(float4*)&sA[wave][r * 20 + cc] = v;
        }
    }
    __syncthreads();

    if (tile < nTiles) {
        v8f c = {0.f, 0.f, 0.f, 0.f, 0.f, 0.f, 0.f, 0.f};
        const int mrow = lane & 15;
        const int koff = (lane >> 4) * 2;
        const int ncol = lane & 15;
        const float* aRow = &sA[wave][mrow * 20];
#pragma unroll
        for (int k = 0; k < 16; k += 4) {
            v2f a, b;
            a.x = aRow[k + koff];
            a.y = aRow[k + koff + 1];
            b.x = sB[(k + koff) * 16 + ncol];
            b.y = sB[(k + koff + 1) * 16 + ncol];
            c = __builtin_amdgcn_wmma_f32_16x16x4_f32(false, a, false, b,
                                                      (short)0, c, false, false);
        }
        if (ncol < 8) {                                  // keep valid 8 columns
            const int row0 = tile * 16 + (lane >> 4) * 8;
#pragma unroll
            for (int j = 0; j < 8; ++j)
                h2[(size_t)(row0 + j) * 8 + ncol] = c[j];
        }
    }
}

// ---------------------------------------------------------------------------
// Per-node attention logits: es = h . a_src, ed = h . a_dst
// ---------------------------------------------------------------------------
template <int F>
__global__ void node_dots(const float* __restrict__ h,
                          const float* __restrict__ a_src,
                          const float* __restrict__ a_dst,
                          float* __restrict__ es, float* __restrict__ ed, int n) {
    int i = blockIdx.x * blockDim.x + threadIdx.x;
    if (i >= n) return;
    const float* hp = h + (size_t)i * F;
    float s0 = 0.f, s1 = 0.f;
#pragma unroll
    for (int f = 0; f < F; ++f) {
        float v = hp[f];
        s0 += v * a_src[f];
        s1 += v * a_dst[f];
    }
    es[i] = s0;
    ed[i] = s1;
}

// ---------------------------------------------------------------------------
// zero accumulators for one layer
// ---------------------------------------------------------------------------
__global__ void init_layer(float* __restrict__ acc, float* __restrict__ s,
                           unsigned* __restrict__ m_u, int n, int F) {
    int i = blockIdx.x * blockDim.x + threadIdx.x;
    if (i < n * F) acc[i] = 0.f;
    if (i < n) { s[i] = 0.f; m_u[i] = 0u; }
}

__device__ __forceinline__ unsigned map_f32(float v) {
    unsigned b = __float_as_uint(v);
    return (b & 0x80000000u) ? ~b : (b | 0x80000000u);
}
__device__ __forceinline__ float unmap_f32(unsigned u) {
    unsigned b = (u & 0x80000000u) ? (u ^ 0x80000000u) : ~u;
    return __uint_as_float(b);
}

// ---------------------------------------------------------------------------
// Edge pass 1: e = leaky_relu(es[src] + ed[dst]); segment-max into m_u[dst]
// Edges e >= nE are implicit self-loops (src=dst=e-nE).
// ---------------------------------------------------------------------------
__global__ void edge_max(const int2* __restrict__ ei,
                         const float* __restrict__ es, const float* __restrict__ ed,
                         float* __restrict__ e_out, unsigned* __restrict__ m_u,
                         int nE, int nTot) {
    int e = blockIdx.x * blockDim.x + threadIdx.x;
    if (e >= nTot) return;
    int src, dst;
    if (e < nE) { int2 p = ei[e]; src = p.x; dst = p.y; }
    else        { src = dst = e - nE; }
    float v = es[src] + ed[dst];
    v = (v > 0.f) ? v : LRELU * v;
    e_out[e] = v;
    atomicMax(&m_u[dst], map_f32(v));
}

// ---------------------------------------------------------------------------
// Edge pass 2: w = exp(e - m[dst]); s[dst]+=w; acc[dst,:] += w*h[src,:]
// ---------------------------------------------------------------------------
template <int F>
__global__ void edge_acc(const int2* __restrict__ ei,
                         const float* __restrict__ e_in,
                         const unsigned* __restrict__ m_u,
                         const float* __restrict__ h,
                         float* __restrict__ s, float* __restrict__ acc,
                         int nE, int nTot) {
    int e = blockIdx.x * blockDim.x + threadIdx.x;
    if (e >= nTot) return;
    int src, dst;
    if (e < nE) { int2 p = ei[e]; src = p.x; dst = p.y; }
    else        { src = dst = e - nE; }
    float w = __expf(e_in[e] - unmap_f32(m_u[dst]));
    atomicAdd(&s[dst], w);
    const float* hp = h + (size_t)src * F;
    float* ap = acc + (size_t)dst * F;
#pragma unroll
    for (int f = 0; f < F; ++f) atomicAdd(&ap[f], w * hp[f]);
}

// ---------------------------------------------------------------------------
// Finalize: out = relu(acc / s + bias)
// ---------------------------------------------------------------------------
template <int F>
__global__ void node_finalize(const float* __restrict__ acc,
                              const float* __restrict__ s,
                              const float* __restrict__ bias,
                              float* __restrict__ out, int n) {
    int i = blockIdx.x * blockDim.x + threadIdx.x;
    if (i >= n * F) return;
    int node = i / F, f = i % F;                  // F is power of 2 -> shifts
    float v = acc[i] / s[node] + bias[f];
    out[i] = (v > 0.f) ? v : 0.f;
}

// ---------------------------------------------------------------------------
// Final linear: out[n] = h2[n,:8] . Wf + bf
// ---------------------------------------------------------------------------
__global__ void final_linear(const float* __restrict__ h,
                             const float* __restrict__ Wf,
                             const float* __restrict__ bf,
                             float* __restrict__ out, int n) {
    int i = blockIdx.x * blockDim.x + threadIdx.x;
    if (i >= n) return;
    const float4* hp = (const float4*)(h + (size_t)i * 8);
    float4 a = hp[0], b = hp[1];
    float r = a.x * Wf[0] + a.y * Wf[1] + a.z * Wf[2] + a.w * Wf[3] +
              b.x * Wf[4] + b.y * Wf[5] + b.z * Wf[6] + b.w * Wf[7] + bf[0];
    out[i] = r;
}

// ---------------------------------------------------------------------------
extern "C" void kernel_launch(void* const* d_in, const int* in_sizes, int n_in,
                              void* d_out, int out_size, void* d_ws, size_t ws_size,
                              hipStream_t stream) {
    const float* x   = (const float*)d_in[0];
    const int2*  ei  = (const int2*)d_in[1];     // int32 pairs (JAX x64 off)
    const float* W1  = (const float*)d_in[2];
    const float* a1s = (const float*)d_in[3];
    const float* a1d = (const float*)d_in[4];
    const float* b1  = (const float*)d_in[5];
    const float* W2  = (const float*)d_in[6];
    const float* a2s = (const float*)d_in[7];
    const float* a2d = (const float*)d_in[8];
    const float* b2  = (const float*)d_in[9];
    const float* Wf  = (const float*)d_in[10];
    const float* bf  = (const float*)d_in[11];
    float* out = (float*)d_out;

    const int N    = in_sizes[0] / 128;          // 100000
    const int E    = in_sizes[1] / 2;            // 3200000
    const int Etot = E + N;

    // Workspace carve-up (floats)
    float* ws = (float*)d_ws;
    float*    h1     = ws;                        // N*16
    float*    outA   = h1 + (size_t)N * 16;       // N*16 (relu of layer-1 out)
    float*    h2     = outA + (size_t)N * 16;     // N*8
    float*    out2   = h2 + (size_t)N * 8;        // N*8 (relu of layer-2 out)
    float*    acc    = out2 + (size_t)N * 8;      // N*16 (shared by both layers)
    float*    s      = acc + (size_t)N * 16;      // N
    unsigned* m_u    = (unsigned*)(s + N);        // N
    float*    es     = (float*)(m_u + N);         // N
    float*    ed     = es + N;                    // N
    float*    e_edge = ed + N;                    // Etot

    const int T = 256;
    const int nTiles   = N / 16;                  // 6250 exact
    const int gemmBlks = (nTiles + 3) / 4;
    const int nodeBlks = (N + T - 1) / T;
    const int nf16Blks = (N * 16 + T - 1) / T;
    const int nf8Blks  = (N * 8 + T - 1) / T;
    const int edgeBlks = (Etot + T - 1) / T;

    // ---------------- Layer 1 (F_IN=128 -> 16) ----------------
    init_layer<<<nf16Blks, T, 0, stream>>>(acc, s, m_u, N, 16);
    gemm_x_w1<<<gemmBlks, 128, 0, stream>>>(x, W1, h1, nTiles);
    node_dots<16><<<nodeBlks, T, 0, stream>>>(h1, a1s, a1d, es, ed, N);
    edge_max<<<edgeBlks, T, 0, stream>>>(ei, es, ed, e_edge, m_u, E, Etot);
    edge_acc<16><<<edgeBlks, T, 0, stream>>>(ei, e_edge, m_u, h1, s, acc, E, Etot);
    node_finalize<16><<<nf16Blks, T, 0, stream>>>(acc, s, b1, outA, N);

    // ---------------- Layer 2 (16 -> 8) ----------------
    init_layer<<<nf16Blks, T, 0, stream>>>(acc, s, m_u, N, 8);
    gemm_h_w2<<<gemmBlks, 128, 0, stream>>>(outA, W2, h2, nTiles);
    node_dots<8><<<nodeBlks, T, 0, stream>>>(h2, a2s, a2d, es, ed, N);
    edge_max<<<edgeBlks, T, 0, stream>>>(ei, es, ed, e_edge, m_u, E, Etot);
    edge_acc<8><<<edgeBlks, T, 0, stream>>>(ei, e_edge, m_u, h2, s, acc, E, Etot);
    node_finalize<8><<<nf8Blks, T, 0, stream>>>(acc, s, b2, out2, N);

    // ---------------- Final linear (8 -> 1) ----------------
    final_linear<<<nodeBlks, T, 0, stream>>>(out2, Wf, bf, out, N);
}